// STEM_v2_50165218018141
// MI455X (gfx1250) — compile-verified
//
#include <hip/hip_runtime.h>

typedef __attribute__((ext_vector_type(2))) float v2f;
typedef __attribute__((ext_vector_type(8))) float v8f;

constexpr int Bc = 32, CIN = 3, Jc = 64, Tc = 256, COUT = 64;
constexpr int SLEN = 5, TLEN = 7;
constexpr int SIGC = 84;                 // 4 + 16 + 64

// workspace (float offsets): folded weight matrices, N-major ([o][k])
constexpr int OFF_AST = 0;                          // [64][84]  spatial  (W_ps_l @ W_ssig, bn_ps-scaled)
constexpr int OFF_ATT = OFF_AST + COUT * SIGC;      // [64][84]  temporal (W_ps_r @ W_tsig, bn_ps-scaled)
constexpr int OFF_WCT = OFF_ATT + COUT * SIGC;      // [64][12]  conv (9 real + 3 zero pad), bn_raw-scaled
constexpr int OFF_WFT = OFF_WCT + COUT * 12;        // [64][128] fuse, bn_fu-scaled
constexpr int OFF_CBS = OFF_WFT + COUT * 128;       // [64] sig  bias (bn folded)
constexpr int OFF_CBR = OFF_CBS + COUT;             // [64] conv bias (bn folded)
constexpr int OFF_CBF = OFF_CBR + COUT;             // [64] fuse bias (bn folded)
constexpr int WSF     = OFF_CBF + COUT;             // 19904 floats total (divisible by 4)

// ---------------------------------------------------------------------------
// Async global -> LDS copy of 16 bytes (CDNA5 GLOBAL_LOAD_ASYNC_TO_LDS_B128,
// tracked by ASYNCcnt). VDST operand carries the LDS byte offset; per the ISA
// flat-aperture rule, the low 32 bits of a generic LDS pointer ARE that offset.
// ---------------------------------------------------------------------------
__device__ __forceinline__ void async_ld16(float* lds_dst, const float* gsrc)
{
    unsigned lds_off = (unsigned)(size_t)lds_dst;
    asm volatile("global_load_async_to_lds_b128 %0, %1, off"
                 :: "v"(lds_off), "v"(gsrc) : "memory");
}
__device__ __forceinline__ void wait_async_all()
{
    asm volatile("s_wait_asynccnt 0x0" ::: "memory");
}

// ---------------------------------------------------------------------------
// Precompute kernel: fold W_ps into W_ssig/W_tsig, fold all BN stages.
// ---------------------------------------------------------------------------
__global__ void stem_precompute(const float* __restrict__ Wss, const float* __restrict__ bss,
                                const float* __restrict__ Wts, const float* __restrict__ bts,
                                const float* __restrict__ Wraw, const float* __restrict__ braw,
                                const float* __restrict__ Wps,  const float* __restrict__ bps,
                                const float* __restrict__ Wfu,  const float* __restrict__ bfu,
                                const float* __restrict__ graw, const float* __restrict__ beraw,
                                const float* __restrict__ mraw, const float* __restrict__ vraw,
                                const float* __restrict__ gps,  const float* __restrict__ beps,
                                const float* __restrict__ mps,  const float* __restrict__ vps,
                                const float* __restrict__ gfu,  const float* __restrict__ befu,
                                const float* __restrict__ mfu,  const float* __restrict__ vfu,
                                float* __restrict__ ws)
{
    const int tid = threadIdx.x;
    // A_s, A_t : [o][kf] = s_ps[o] * sum_i W_ps[o][i(+64)] * W_{s,t}sig[i][kf]
    for (int p = tid; p < COUT * SIGC; p += blockDim.x) {
        const int o = p / SIGC, kf = p % SIGC;
        const float s = gps[o] * rsqrtf(vps[o] + 1e-5f);
        float as = 0.f, at = 0.f;
        for (int i = 0; i < COUT; ++i) {
            as += Wps[o * 128 + i]      * Wss[i * SIGC + kf];
            at += Wps[o * 128 + 64 + i] * Wts[i * SIGC + kf];
        }
        ws[OFF_AST + p] = as * s;
        ws[OFF_ATT + p] = at * s;
    }
    // conv weights, K padded 9 -> 12 (multiple of 4 for WMMA K-steps)
    for (int p = tid; p < COUT * 12; p += blockDim.x) {
        const int o = p / 12, kk = p % 12;
        const float s = graw[o] * rsqrtf(vraw[o] + 1e-5f);
        ws[OFF_WCT + p] = (kk < 9) ? Wraw[o * 9 + kk] * s : 0.f;
    }
    // fuse weights
    for (int p = tid; p < COUT * 128; p += blockDim.x) {
        const int o = p / 128;
        const float s = gfu[o] * rsqrtf(vfu[o] + 1e-5f);
        ws[OFF_WFT + p] = Wfu[p] * s;
    }
    // folded biases
    for (int o = tid; o < COUT; o += blockDim.x) {
        const float sps  = gps[o]  * rsqrtf(vps[o]  + 1e-5f);
        const float sraw = graw[o] * rsqrtf(vraw[o] + 1e-5f);
        const float sfu  = gfu[o]  * rsqrtf(vfu[o]  + 1e-5f);
        float pre = bps[o];
        for (int i = 0; i < COUT; ++i)
            pre += Wps[o * 128 + i] * bss[i] + Wps[o * 128 + 64 + i] * bts[i];
        ws[OFF_CBS + o] = (pre - mps[o]) * sps + beps[o];
        ws[OFF_CBR + o] = (braw[o] - mraw[o]) * sraw + beraw[o];
        ws[OFF_CBF + o] = (bfu[o]  - mfu[o])  * sfu  + befu[o];
    }
}

// ---------------------------------------------------------------------------
// Depth-3 signature with basepoint for an L-point path of 4 channels.
// Fully unrolled -> state stays in VGPRs; features written straight to LDS.
// Feature order matches reference concat: [S1(4), S2(16), S3(64)].
// ---------------------------------------------------------------------------
template <int L>
__device__ __forceinline__ void sig3_to(const float* __restrict__ pts /*[L][4]*/,
                                        float* __restrict__ feat /*LDS, 84*/)
{
    float S1[4] = {}, S2[16] = {}, S3[64] = {}, prev[4] = {};
#pragma unroll
    for (int l = 0; l < L; ++l) {
        float d[4];
#pragma unroll
        for (int i = 0; i < 4; ++i) { d[i] = pts[l * 4 + i] - prev[i]; prev[i] = pts[l * 4 + i]; }
        float e2[16];
#pragma unroll
        for (int i = 0; i < 4; ++i)
#pragma unroll
            for (int j = 0; j < 4; ++j) e2[i * 4 + j] = 0.5f * d[i] * d[j];
        float f[4];
#pragma unroll
        for (int i = 0; i < 4; ++i) f[i] = S1[i] + d[i] * (1.f / 3.f);
        // S3 += S2[:,:,None]*d + (S1 + d/3)[:,None,None]*e2   (uses OLD S2,S1)
#pragma unroll
        for (int i = 0; i < 4; ++i)
#pragma unroll
            for (int j = 0; j < 4; ++j)
#pragma unroll
                for (int k = 0; k < 4; ++k)
                    S3[i * 16 + j * 4 + k] += S2[i * 4 + j] * d[k] + f[i] * e2[j * 4 + k];
#pragma unroll
        for (int i = 0; i < 4; ++i)
#pragma unroll
            for (int j = 0; j < 4; ++j) S2[i * 4 + j] += S1[i] * d[j] + e2[i * 4 + j];
#pragma unroll
        for (int i = 0; i < 4; ++i) S1[i] += d[i];
    }
#pragma unroll
    for (int i = 0; i < 4; ++i)  feat[i]      = S1[i];
#pragma unroll
    for (int i = 0; i < 16; ++i) feat[4 + i]  = S2[i];
#pragma unroll
    for (int i = 0; i < 64; ++i) feat[20 + i] = S3[i];
}

// ---------------------------------------------------------------------------
// Wave-level f32 WMMA: D(16pt x 16out) += A(16x4) x B(4x16), K stepped.
// A: LDS scratch rows [32][128]; B: weights stored N-major [64][BSTRIDE].
// Per ISA 32-bit layouts: row/col = lane&15, K-half = lane>>4, consecutive
// K pair -> one ds_load_b64 for both A and B fragments.
// ---------------------------------------------------------------------------
template <int KQ, int BSTRIDE>
__device__ __forceinline__ void wmma_mm(const float* __restrict__ aw,
                                        const float* __restrict__ bw,
                                        int col, int kh, v8f acc[4][2])
{
#pragma unroll
    for (int n = 0; n < 4; ++n) {
#pragma unroll
        for (int m = 0; m < 2; ++m) {
            const float* ap = aw + (m * 16 + col) * 128;
            const float* bp = bw + (n * 16 + col) * BSTRIDE;
            v8f c = acc[n][m];
#pragma unroll
            for (int k = 0; k < KQ; ++k) {
                v2f a = *(const v2f*)(ap + k * 4 + 2 * kh);
                v2f b = *(const v2f*)(bp + k * 4 + 2 * kh);
                c = __builtin_amdgcn_wmma_f32_16x16x4_f32(false, a, false, b,
                                                          (short)0, c, false, false);
            }
            acc[n][m] = c;
        }
    }
}

// ---------------------------------------------------------------------------
// Main fused kernel: one workgroup per (b, j); 8 wave32s, 32 t-points/wave.
// ---------------------------------------------------------------------------
__launch_bounds__(256)
__global__ void stem_main(const float* __restrict__ x, const int* __restrict__ path,
                          const float* __restrict__ ws, float* __restrict__ out)
{
    __shared__ float smem[57280];                    // 229,120 B (< 320 KB WGP LDS)
    float* xrow    = smem;                           // [3][256]
    float* xpath   = smem + 768;                     // [5][3][256]
    float* w       = smem + 4608;                    // WSF = 19904 folded weights
    float* scratch = smem + 24512;                   // [8 waves][32 rows][128]

    const int tid = threadIdx.x;
    const int b   = blockIdx.x >> 6;
    const int j   = blockIdx.x & 63;

    const float* xb = x + (size_t)b * CIN * Jc * Tc;

    // ---- stage weights + input rows into LDS via async B128 copies ----
    for (int q = tid; q < WSF / 4; q += 256)         // 4976 x float4
        async_ld16(w + q * 4, ws + q * 4);
    for (int q = tid; q < (CIN * Tc) / 4; q += 256) { // 192 x float4
        const int c = q >> 6, t4 = q & 63;
        async_ld16(xrow + q * 4, xb + (c * Jc + j) * Tc + t4 * 4);
    }
    for (int q = tid; q < (SLEN * CIN * Tc) / 4; q += 256) { // 960 x float4
        const int l = q / 192, c = (q >> 6) % 3, t4 = q & 63;
        const int jj = path[j * SLEN + l];
        async_ld16(xpath + q * 4, xb + (c * Jc + jj) * Tc + t4 * 4);
    }
    wait_async_all();                                // drain this wave's ASYNCcnt
    __syncthreads();                                 // all waves' copies visible

    const int wave = tid >> 5;
    const int lane = tid & 31;
    const int col  = lane & 15;
    const int kh   = lane >> 4;
    const int tpt  = wave * 32 + lane;               // this lane's t
    float* sw = scratch + wave * 32 * 128;           // wave-private scratch

    v8f accS[4][2] = {};                             // fused sig accumulators

    // ---- temporal signature features (edge-clamped window of 7) ----
    {
        float pts[TLEN * 4];
#pragma unroll
        for (int k = 0; k < TLEN; ++k) {
            int tt = tpt + k - 3;
            tt = tt < 0 ? 0 : (tt > Tc - 1 ? Tc - 1 : tt);
#pragma unroll
            for (int c = 0; c < 3; ++c) pts[k * 4 + c] = xrow[c * Tc + tt];
            pts[k * 4 + 3] = (float)k * (1.f / 6.f);
        }
        sig3_to<TLEN>(pts, sw + lane * 128);
    }
    __syncthreads();
    wmma_mm<21, SIGC>(sw, w + OFF_ATT, col, kh, accS);   // += A_t @ F_t
    __syncthreads();

    // ---- spatial signature features (5 joints along path[j]) ----
    {
        float pts[SLEN * 4];
#pragma unroll
        for (int l = 0; l < SLEN; ++l) {
#pragma unroll
            for (int c = 0; c < 3; ++c) pts[l * 4 + c] = xpath[(l * 3 + c) * Tc + tpt];
            pts[l * 4 + 3] = (float)l * 0.25f;
        }
        sig3_to<SLEN>(pts, sw + lane * 128);
    }
    __syncthreads();
    wmma_mm<21, SIGC>(sw, w + OFF_AST, col, kh, accS);   // += A_s @ F_s
    __syncthreads();

    // ---- 1x3 conv features (zero-padded), K padded to 12 ----
    {
        float* fr = sw + lane * 128;
#pragma unroll
        for (int c = 0; c < 3; ++c)
#pragma unroll
            for (int k = 0; k < 3; ++k) {
                const int tt = tpt + k - 1;
                fr[c * 3 + k] = (tt >= 0 && tt < Tc) ? xrow[c * Tc + tt] : 0.f;
            }
        fr[9] = 0.f; fr[10] = 0.f; fr[11] = 0.f;
    }
    __syncthreads();
    v8f accC[4][2] = {};
    wmma_mm<3, 12>(sw, w + OFF_WCT, col, kh, accC);
    __syncthreads();

    // ---- finalize sig & conv: bias + ReLU, build concat [x_conv | sig] ----
    const float* cbS = w + OFF_CBS;
    const float* cbR = w + OFF_CBR;
    const float* cbF = w + OFF_CBF;
#pragma unroll
    for (int n = 0; n < 4; ++n) {
        const int o = n * 16 + col;
        const float bs = cbS[o], br = cbR[o];
#pragma unroll
        for (int m = 0; m < 2; ++m) {
            const v8f cs = accS[n][m];
            const v8f cc = accC[n][m];
#pragma unroll
            for (int v = 0; v < 8; ++v) {
                const int row = m * 16 + v + 8 * kh;     // C-layout: M = v (+8 for hi lanes)
                sw[row * 128 + o]       = fmaxf(cc[v] + br, 0.f);   // x_conv -> cols 0..63
                sw[row * 128 + 64 + o]  = fmaxf(cs[v] + bs, 0.f);   // sig    -> cols 64..127
            }
        }
    }
    __syncthreads();

    // ---- fuse: out = relu( W_fu' @ [x_conv ; sig] + cb_fu ), K = 128 ----
    v8f accO[4][2] = {};
    wmma_mm<32, 128>(sw, w + OFF_WFT, col, kh, accO);

    // ---- store: coalesced float4 along t ----
#pragma unroll
    for (int n = 0; n < 4; ++n) {
        const int o = n * 16 + col;
        const float bf = cbF[o];
        const size_t base = (((size_t)b * COUT + o) * Jc + j) * Tc;
#pragma unroll
        for (int m = 0; m < 2; ++m) {
            const v8f c = accO[n][m];
            const int t0 = wave * 32 + m * 16 + 8 * kh;  // v=0..3 -> t0..t0+3, v=4..7 -> t0+4..7
            float4 lo, hi;
            lo.x = fmaxf(c[0] + bf, 0.f); lo.y = fmaxf(c[1] + bf, 0.f);
            lo.z = fmaxf(c[2] + bf, 0.f); lo.w = fmaxf(c[3] + bf, 0.f);
            hi.x = fmaxf(c[4] + bf, 0.f); hi.y = fmaxf(c[5] + bf, 0.f);
            hi.z = fmaxf(c[6] + bf, 0.f); hi.w = fmaxf(c[7] + bf, 0.f);
            *(float4*)(out + base + t0)     = lo;
            *(float4*)(out + base + t0 + 4) = hi;
        }
    }
}

// ---------------------------------------------------------------------------
extern "C" void kernel_launch(void* const* d_in, const int* in_sizes, int n_in,
                              void* d_out, int out_size, void* d_ws, size_t ws_size,
                              hipStream_t stream)
{
    (void)in_sizes; (void)n_in; (void)out_size; (void)ws_size;
    const float* x   = (const float*)d_in[0];
    const int* path  = (const int*)d_in[1];
    float* ws        = (float*)d_ws;

    stem_precompute<<<1, 256, 0, stream>>>(
        (const float*)d_in[2],  (const float*)d_in[3],   // W_ssig, b_ssig
        (const float*)d_in[4],  (const float*)d_in[5],   // W_tsig, b_tsig
        (const float*)d_in[6],  (const float*)d_in[7],   // W_raw,  b_raw
        (const float*)d_in[8],  (const float*)d_in[9],   // W_ps,   b_ps
        (const float*)d_in[10], (const float*)d_in[11],  // W_fu,   b_fu
        (const float*)d_in[12], (const float*)d_in[13], (const float*)d_in[14], (const float*)d_in[15], // raw bn
        (const float*)d_in[16], (const float*)d_in[17], (const float*)d_in[18], (const float*)d_in[19], // ps  bn
        (const float*)d_in[20], (const float*)d_in[21], (const float*)d_in[22], (const float*)d_in[23], // fu  bn
        ws);

    stem_main<<<Bc * Jc, 256, 0, stream>>>(x, path, ws, (float*)d_out);
}